// MultiHeadAttention_25632364822723
// MI455X (gfx1250) — compile-verified
//
#include <hip/hip_runtime.h>

typedef float v2f __attribute__((ext_vector_type(2)));
typedef float v8f __attribute__((ext_vector_type(8)));

#define B_ 16
#define S_ 2048
#define D_ 128
#define NEG_BIG (-1.0e9f)
#define QK_SCALE 0.0883883476483184f /* 1/sqrt(128) */
#define STR 36 /* padded LDS row stride (floats): bank step 36 -> conflict-free b64 frags */

// One block = (batch b, 16-row query strip). 256 threads = 8 waves (wave32).
// Wave w owns key range [w*256, (w+1)*256) in phases 1/2a and d-tile [w*16, w*16+16) in 2b.
__global__ __launch_bounds__(256) void mha_fused_kernel(
    const float* __restrict__ Q, const float* __restrict__ K,
    const float* __restrict__ V, const float* __restrict__ Mk,
    float* __restrict__ O, float* __restrict__ W)
{
  __shared__ float stage[8][2][16 * STR];   // 36.9 KB: two padded 16x32 tiles per wave
  __shared__ float lds_m[8][16];
  __shared__ float lds_s[8][16];
  __shared__ float lds_M[16];
  __shared__ float lds_invS[16];

  const int tid   = threadIdx.x;
  const int wave  = tid >> 5;
  const int lane  = tid & 31;
  const int row16 = lane & 15;
  const int half  = lane >> 4;
  const int rbase = half * 8;

  const int b  = blockIdx.x >> 7;
  const int q0 = (blockIdx.x & 127) << 4;

  const float* Qb = Q  + ((size_t)b * S_ + q0) * D_;
  const float* Kb = K  + (size_t)b * S_ * D_;
  const float* Vb = V  + (size_t)b * S_ * D_;
  const float* Mb = Mk + ((size_t)b * S_ + q0) * S_;
  float*       Wb = W  + ((size_t)b * S_ + q0) * S_;
  float*       Ob = O  + ((size_t)b * S_ + q0) * D_;

  float* buf0 = &stage[wave][0][0];
  float* buf1 = &stage[wave][1][0];

  // Coalesced 16x32 f32 tile: global (row stride rs) -> 8 float2 per lane.
  // Per instr: each half-wave covers a contiguous 128B run of one row.
  auto load16x32 = [&](v2f* r, const float* src, int rs) {
#pragma unroll
    for (int i = 0; i < 8; ++i)
      r[i] = *(const v2f*)(src + (size_t)(2 * i + half) * rs + 2 * row16);
  };
  // Staged tile -> padded LDS (row stride STR floats).
  auto store16x32 = [&](float* dst, const v2f* r) {
#pragma unroll
    for (int i = 0; i < 8; ++i)
      *(v2f*)(dst + (2 * i + half) * STR + 2 * row16) = r[i];
  };

  const v8f vzero = {};

  // ---------------- Phase 1: masked logits (WMMA) + online (max, sumexp) ----------------
  v2f afrag[32];   // Q tile, pre-scaled; all indices constant -> stays in VGPRs
#pragma unroll
  for (int s = 0; s < 32; ++s) {
    v2f a = *(const v2f*)(Qb + (size_t)row16 * D_ + s * 4 + half * 2);
    afrag[s].x = a.x * QK_SCALE;
    afrag[s].y = a.y * QK_SCALE;
  }

  float mrun[8], srun[8];
#pragma unroll
  for (int r = 0; r < 8; ++r) { mrun[r] = -__builtin_inff(); srun[r] = 0.0f; }

  // Key tile tt (16 keys) x feature chunk ch (32 feats). Ping-pong parity == ch&1.
  v2f sreg[8];
  load16x32(sreg, Kb + (size_t)((wave * 16) * 16) * D_, D_);
  store16x32(buf0, sreg);
  for (int tt = 0; tt < 16; ++tt) {       // dynamic outer loop
    const float* Kt = Kb + (size_t)((wave * 16 + tt) * 16) * D_;
    v8f c = vzero;
    if (tt < 14) {  // prefetch mask stream two key-tiles ahead (one line per row)
      const int kpf = (wave * 16 + tt + 2) * 16;
      __builtin_prefetch(Mb + (size_t)row16 * S_ + kpf, 0, 1);
    }
#pragma unroll
    for (int ch = 0; ch < 4; ++ch) {      // fully unrolled: constant afrag index + buffer parity
      // software pipeline: issue next chunk's globals before consuming this one
      if (ch < 3)            load16x32(sreg, Kt + (ch + 1) * 32, D_);
      else if (tt < 15)      load16x32(sreg, Kt + (size_t)16 * D_, D_);
      const float* cb = (ch & 1) ? buf1 : buf0;
#pragma unroll
      for (int s = 0; s < 8; ++s) {
        v2f bf = *(const v2f*)(cb + row16 * STR + s * 4 + 2 * half);  // conflict-free ds_load_b64
        c = __builtin_amdgcn_wmma_f32_16x16x4_f32(false, afrag[ch * 8 + s],
                                                  false, bf, (short)0, c, false, false);
      }
      if (ch < 3 || tt < 15) store16x32((ch & 1) ? buf0 : buf1, sreg);  // DS in-order: no WAR
    }
    // tile epilogue: mask, store raw logits, online softmax stats
    const int kc0 = (wave * 16 + tt) * 16;
#pragma unroll
    for (int r = 0; r < 8; ++r) {
      const int rr = r + rbase;
      const size_t idx = (size_t)rr * S_ + kc0 + row16;
      float x = c[r] + NEG_BIG * Mb[idx];
      Wb[idx] = x;
      float mo = mrun[r];
      float mn = fmaxf(mo, x);
      srun[r] = srun[r] * __expf(mo - mn) + __expf(x - mn);
      mrun[r] = mn;
    }
  }

  // ---------------- Softmax reduction: 16-lane halves, then across waves via LDS ----------------
#pragma unroll
  for (int r = 0; r < 8; ++r) {
#pragma unroll
    for (int off = 1; off < 16; off <<= 1) {
      float mo2 = __shfl_xor(mrun[r], off, 32);
      float so2 = __shfl_xor(srun[r], off, 32);
      float mn  = fmaxf(mrun[r], mo2);
      srun[r] = srun[r] * __expf(mrun[r] - mn) + so2 * __expf(mo2 - mn);
      mrun[r] = mn;
    }
  }
  if (row16 == 0) {
#pragma unroll
    for (int r = 0; r < 8; ++r) {
      lds_m[wave][r + rbase] = mrun[r];
      lds_s[wave][r + rbase] = srun[r];
    }
  }
  __syncthreads();
  if (tid < 16) {
    float M = -__builtin_inff(), Ssum = 0.0f;
#pragma unroll
    for (int w = 0; w < 8; ++w) {
      float m2 = lds_m[w][tid], s2 = lds_s[w][tid];
      float mn = fmaxf(M, m2);
      Ssum = Ssum * __expf(M - mn) + s2 * __expf(m2 - mn);
      M = mn;
    }
    lds_M[tid]    = M;
    lds_invS[tid] = 1.0f / Ssum;
  }
  __syncthreads();

  // ---------------- Phase 2a: normalize own columns (same-wave store->load ordered) ----------------
  {
    const int cbase = wave * 256;
    for (int i = lane; i < 4096; i += 32) {
      const int r = i >> 8;
      const size_t idx = (size_t)r * S_ + cbase + (i & 255);
      float x = Wb[idx];
      Wb[idx] = __expf(x - lds_M[r]) * lds_invS[r];
    }
  }
  __threadfence();   // publish normalized weights across waves
  __syncthreads();

  // ---------------- Phase 2b: O = P * V via WMMA, LDS-staged both streams ----------------
  {
    const int d0 = wave * 16;
    v2f  rA[8];
    float rV[16];
    auto loadA = [&](int g) { load16x32(rA, Wb + g * 32, S_); };
    auto loadV = [&](int g) {   // V chunk, coalesced 64B per half-wave
      const float* vs = Vb + (size_t)(g * 32) * D_ + d0;
#pragma unroll
      for (int i = 0; i < 16; ++i)
        rV[i] = vs[(size_t)(2 * i + half) * D_ + row16];
    };
    loadA(0);
    loadV(0);
    v8f o = vzero;
    for (int g = 0; g < 64; ++g) {
      store16x32(buf0, rA);            // P chunk: [q-row][key] padded
#pragma unroll
      for (int i = 0; i < 16; ++i)     // V chunk transposed: [feat][key] padded
        buf1[row16 * STR + 2 * i + half] = rV[i];
      if (g < 63) { loadA(g + 1); loadV(g + 1); }  // next chunk's globals in flight
#pragma unroll
      for (int s = 0; s < 8; ++s) {
        v2f af = *(const v2f*)(buf0 + row16 * STR + s * 4 + 2 * half);
        v2f bf = *(const v2f*)(buf1 + row16 * STR + s * 4 + 2 * half);
        o = __builtin_amdgcn_wmma_f32_16x16x4_f32(false, af, false, bf,
                                                  (short)0, o, false, false);
      }
    }
#pragma unroll
    for (int r = 0; r < 8; ++r)
      Ob[(size_t)(r + rbase) * D_ + d0 + row16] = o[r];
  }
}

extern "C" void kernel_launch(void* const* d_in, const int* in_sizes, int n_in,
                              void* d_out, int out_size, void* d_ws, size_t ws_size,
                              hipStream_t stream) {
  (void)in_sizes; (void)n_in; (void)d_ws; (void)ws_size; (void)out_size;
  const float* Q = (const float*)d_in[0];
  const float* K = (const float*)d_in[1];
  const float* V = (const float*)d_in[2];
  const float* M = (const float*)d_in[3];
  float* O = (float*)d_out;                    // [B,S,D] first output
  float* W = O + (size_t)B_ * S_ * D_;         // [B,S,S] second output
  dim3 grid(B_ * (S_ / 16));                   // 2048 blocks
  dim3 block(256);                             // 8 waves
  hipLaunchKernelGGL(mha_fused_kernel, grid, block, 0, stream, Q, K, V, M, O, W);
}